// LastShared_88467736363910
// MI455X (gfx1250) — compile-verified
//
#include <hip/hip_runtime.h>

typedef float v2f __attribute__((ext_vector_type(2)));
typedef float v8f __attribute__((ext_vector_type(8)));

// ---------------------------------------------------------------------------
// Zero-fill (float4 granularity; n must be a multiple of 4)
// ---------------------------------------------------------------------------
__global__ void zero_f32v4(float4* __restrict__ p, long long n4) {
    long long i = (long long)blockIdx.x * blockDim.x + threadIdx.x;
    if (i < n4) p[i] = make_float4(0.f, 0.f, 0.f, 0.f);
}

// ---------------------------------------------------------------------------
// Dense GEMM via V_WMMA_F32_16X16X4_F32 (exact fp32 matrix path on CDNA5).
// C[M,NN] = op(A[M,KK]) @ B[KK,NN], op = relu if RELU_A.
//
// Block = 256 threads = 8 waves. All waves share one N-tile (blockIdx.y);
// the 16-wide B panel is staged once into LDS, transposed with +4 float row
// pad (row stride 516/132 -> stride mod 64banks = 4 -> conflict-free
// ds_load_b64 for both lane halves). Each wave owns one 16x16 output tile.
// NN, KK are compile-time so all addressing is 32-bit with immediate offsets.
// ---------------------------------------------------------------------------
template <int RELU_A, int NN, int KK>
__global__ void gemm_wmma_f32(const float* __restrict__ A,
                              const float* __restrict__ B,
                              float* __restrict__ C,
                              int M) {
    constexpr int LDB = KK + 4;                 // padded LDS row stride (floats)
    __shared__ float Bt[16 * LDB];              // Bt[n][k] = B[k][n0+n]

    const int tid  = (int)threadIdx.x;
    const int n0   = (int)blockIdx.y << 4;

    // Cooperative load of the B panel: 16 consecutive n per k-row (coalesced),
    // transposed store into LDS (conflict-free: lanes differ in n -> bank 4n+k).
    #pragma unroll
    for (int idx = 0; idx < KK * 16 / 256; ++idx) {
        int flat = tid + idx * 256;
        int k = flat >> 4;
        int n = flat & 15;
        Bt[n * LDB + k] = B[k * NN + n0 + n];
    }
    __syncthreads();

    const int wave = tid >> 5;
    const int lane = tid & 31;
    const int tileM = (int)blockIdx.x * 8 + wave;
    if (tileM >= (M >> 4)) return;              // wave-uniform exit

    const int l15  = lane & 15;
    const int kOff = (lane >> 4) << 1;          // 0 for lanes 0-15, 2 for 16-31

    const float* __restrict__ arow = A + ((tileM << 4) + l15) * KK + kOff;
    const float* __restrict__ brow = &Bt[l15 * LDB + kOff];

    v8f acc = {};
    #pragma unroll 16
    for (int k0 = 0; k0 < KK; k0 += 4) {
        float2 av = *(const float2*)(arow + k0);     // global_load_b64, imm offset
        if (RELU_A) {
            av.x = fmaxf(av.x, 0.f);
            av.y = fmaxf(av.y, 0.f);
        }
        v2f a; a.x = av.x; a.y = av.y;
        v2f b = *(const v2f*)(brow + k0);            // ds_load_b64, imm offset
        acc = __builtin_amdgcn_wmma_f32_16x16x4_f32(
            /*neg_a=*/false, a, /*neg_b=*/false, b,
            /*c_mod=*/(short)0, acc, /*reuse_a=*/false, /*reuse_b=*/false);
    }

    // D layout: VGPR r -> M = r + 8*(lane>=16), N = lane&15
    const int cm = (tileM << 4) + ((lane >> 4) << 3);
    float* __restrict__ crow = C + cm * NN + n0 + l15;
    #pragma unroll
    for (int r = 0; r < 8; ++r) crow[r * NN] = acc[r];
}

// ---------------------------------------------------------------------------
// Edge-parallel SpMM scatter: out[row[e], :] += w[e] * h[col[e], :].
// C = 4 << SHIFT channels; (1<<SHIFT) lanes cooperate per edge, each lane
// owns a float4 channel chunk. Gathers are L2-resident (h tables fit in the
// 192MB L2); scatter uses global_atomic_add_f32.
// ---------------------------------------------------------------------------
template <int SHIFT>
__global__ void spmm_atomic(const int*   __restrict__ erow,
                            const int*   __restrict__ ecol,
                            const float* __restrict__ ew,
                            const float* __restrict__ h,
                            float*       __restrict__ out,
                            int nEdges) {
    const long long tid = (long long)blockIdx.x * blockDim.x + threadIdx.x;
    const int e  = (int)(tid >> SHIFT);
    if (e >= nEdges) return;
    const int c4 = ((int)tid & ((1 << SHIFT) - 1)) << 2;
    constexpr int Cch = 4 << SHIFT;

    const float we = ew[e];
    const int   j  = ecol[e];
    const int   i  = erow[e];

    const float4 hv = *(const float4*)(h + (size_t)j * Cch + c4);
    float* dst = out + (size_t)i * Cch + c4;
    atomicAdd(dst + 0, hv.x * we);
    atomicAdd(dst + 1, hv.y * we);
    atomicAdd(dst + 2, hv.z * we);
    atomicAdd(dst + 3, hv.w * we);
}

// ---------------------------------------------------------------------------
// Launcher
// ---------------------------------------------------------------------------
extern "C" void kernel_launch(void* const* d_in, const int* in_sizes, int n_in,
                              void* d_out, int out_size, void* d_ws, size_t ws_size,
                              hipStream_t stream) {
    const float* x   = (const float*)d_in[0];   // [N, 512]
    const float* W1  = (const float*)d_in[1];   // [512, 128]
    const float* W2  = (const float*)d_in[2];   // [128, 64]
    const float* ew  = (const float*)d_in[3];   // [E]
    const int*   er  = (const int*)  d_in[4];   // [E]
    const int*   ec  = (const int*)  d_in[5];   // [E]

    const int F_IN = 512, C1 = 128, C2 = 64;
    const int nNodes = in_sizes[0] / F_IN;      // 50000
    const int nEdges = in_sizes[3];             // 1.6M

    float* H   = (float*)d_ws;                        // [N, 128]
    float* Hs  = H  + (size_t)nNodes * C1;            // [N, 128]
    float* H2  = Hs + (size_t)nNodes * C1;            // [N, 64]
    float* out = (float*)d_out;                       // [N, 64]

    const int BLK = 256;

    // Zero accumulation targets (harness poisons d_out / d_ws).
    {
        long long n4 = (long long)nNodes * C1 / 4;
        zero_f32v4<<<(unsigned)((n4 + BLK - 1) / BLK), BLK, 0, stream>>>((float4*)Hs, n4);
        long long m4 = (long long)nNodes * C2 / 4;
        zero_f32v4<<<(unsigned)((m4 + BLK - 1) / BLK), BLK, 0, stream>>>((float4*)out, m4);
    }

    // GEMM1: H = x @ W1   (fp32 WMMA, B panel in LDS)
    {
        int mTiles = nNodes / 16;                       // 3125
        dim3 grid((mTiles + 7) / 8, C1 / 16);           // (391, 8)
        gemm_wmma_f32<0, 128, 512><<<grid, BLK, 0, stream>>>(x, W1, H, nNodes);
    }

    // SpMM1: Hs = A @ H   (C=128 -> 32 lanes/edge)
    {
        long long threads = (long long)nEdges << 5;
        spmm_atomic<5><<<(unsigned)((threads + BLK - 1) / BLK), BLK, 0, stream>>>(
            er, ec, ew, H, Hs, nEdges);
    }

    // GEMM2: H2 = relu(Hs) @ W2   (relu fused into A-read)
    {
        int mTiles = nNodes / 16;
        dim3 grid((mTiles + 7) / 8, C2 / 16);           // (391, 4)
        gemm_wmma_f32<1, 64, 128><<<grid, BLK, 0, stream>>>(Hs, W2, H2, nNodes);
    }

    // SpMM2: out = A @ H2  (C=64 -> 16 lanes/edge)
    {
        long long threads = (long long)nEdges << 4;
        spmm_atomic<4><<<(unsigned)((threads + BLK - 1) / BLK), BLK, 0, stream>>>(
            er, ec, ew, H2, out, nEdges);
    }
}